// SemiSparseCRFLoss_71201967833119
// MI455X (gfx1250) — compile-verified
//
#include <hip/hip_runtime.h>
#include <hip/hip_bf16.h>

// Problem geometry (fixed by the reference).
#define HH 512
#define WW 512
#define BB 16
#define ROWS 4          // rows of output per block (plus 1 halo row staged)
#define THREADS 256     // 8 wave32 per block; 64 threads/row, 8 px/thread

#ifndef __has_builtin
#define __has_builtin(x) 0
#endif

#if __has_builtin(__builtin_amdgcn_global_load_async_to_lds_b128)
#define USE_ASYNC_LDS 1
#else
#define USE_ASYNC_LDS 0
#endif

typedef int v4i __attribute__((ext_vector_type(4)));
typedef __attribute__((address_space(1))) v4i gm_v4i;   // AS1: global ("__device__")
typedef __attribute__((address_space(3))) v4i lds_v4i;  // AS3: LDS ("__shared__")

__device__ __forceinline__ float fast_exp2(float x) {
#if __has_builtin(__builtin_amdgcn_exp2f)
    return __builtin_amdgcn_exp2f(x);   // v_exp_f32 directly
#else
    return exp2f(x);                    // OCML native exp2 -> v_exp_f32
#endif
}

__device__ __forceinline__ void stage16(void* lds_dst, const void* g_src) {
#if USE_ASYNC_LDS
    // gfx1250 async LDS-DMA: per-lane global addr -> per-lane LDS addr, ASYNCcnt-tracked.
    __builtin_amdgcn_global_load_async_to_lds_b128((gm_v4i*)g_src, (lds_v4i*)lds_dst,
                                                   /*offset=*/0, /*cpol=*/0);
#else
    *(float4*)lds_dst = *(const float4*)g_src;
#endif
}

__device__ __forceinline__ void drain_async() {
#if USE_ASYNC_LDS
#if __has_builtin(__builtin_amdgcn_s_wait_asynccnt)
    __builtin_amdgcn_s_wait_asynccnt(0);
#else
    asm volatile("s_wait_asynccnt 0" ::: "memory");
#endif
#endif
}

__global__ void crf_zero_kernel(float* out) {
    if (threadIdx.x == 0) out[0] = 0.0f;
}

__global__ __launch_bounds__(THREADS)
void crf_loss_kernel(const float* __restrict__ y_pr,
                     const float* __restrict__ image,
                     const int*   __restrict__ image_class,
                     float*       __restrict__ out) {
    // 4 channels (R,G,B,y1), ROWS+1 rows of 512 floats each: 40 KB.
    __shared__ __align__(16) float s_ch[4][(ROWS + 1) * WW];
    __shared__ float s_red[8];

    const int bid  = blockIdx.x;
    const int b    = bid >> 7;          // / 128 tiles per image
    const int tile = bid & 127;
    const int h0   = tile * ROWS;

    // sel = (image_class != NEG_CLASS); skip the whole strip (loads included) when gated.
    if (image_class[b] == 0) return;

    const int tid   = threadIdx.x;
    const int nrows = (h0 + ROWS < HH) ? (ROWS + 1) : ROWS;   // stage halo row if it exists
    const int nvec  = nrows * (WW / 4);                        // float4 per channel

    // Rows h0..h0+nrows-1 of one channel are CONTIGUOUS in memory.
    const float* src0 = image + ((size_t)(b * 3 + 0) * HH + h0) * WW;
    const float* src1 = image + ((size_t)(b * 3 + 1) * HH + h0) * WW;
    const float* src2 = image + ((size_t)(b * 3 + 2) * HH + h0) * WW;
    const float* src3 = y_pr  + ((size_t)(b * 2 + 1) * HH + h0) * WW;  // softmax channel 1
    const float* srcs[4] = {src0, src1, src2, src3};

    #pragma unroll
    for (int c = 0; c < 4; ++c) {
        const float4* g = (const float4*)srcs[c];
        float4*       l = (float4*)&s_ch[c][0];
        for (int i = tid; i < nvec; i += THREADS)
            stage16(&l[i], &g[i]);
    }
    drain_async();      // my async copies have landed in LDS
    __syncthreads();    // everyone's have

    // exp(-d2/(2*0.15^2)) == exp2(d2 * -22.2222*log2(e)); w_xy = e^-1/2, e^-1.
    const float kExp2Scale = -32.0598895416f;
    const float kW10 = 0.60653065971263342f;
    const float kW11 = 0.36787944117144233f;

    // Each thread owns 8 contiguous pixels of one row: registers carry all taps.
    const int r     = tid >> 6;          // 0..3
    const int w0    = (tid & 63) << 3;   // 0,8,...,504
    const int h     = h0 + r;
    const bool below = (h + 1 < HH);
    const int o     = r * WW + w0;

    float cur[4][9];    // row r,  cols w0   .. w0+8
    float dn [4][10];   // row r+1, cols w0-1 .. w0+8

    #pragma unroll
    for (int c = 0; c < 4; ++c) {
        float4 a = *(const float4*)&s_ch[c][o];
        float4 bq = *(const float4*)&s_ch[c][o + 4];
        cur[c][0] = a.x; cur[c][1] = a.y; cur[c][2] = a.z; cur[c][3] = a.w;
        cur[c][4] = bq.x; cur[c][5] = bq.y; cur[c][6] = bq.z; cur[c][7] = bq.w;
        cur[c][8] = s_ch[c][o + 8];                       // validity enforced at use
        if (below) {
            float4 e = *(const float4*)&s_ch[c][o + WW];
            float4 f = *(const float4*)&s_ch[c][o + WW + 4];
            dn[c][1] = e.x; dn[c][2] = e.y; dn[c][3] = e.z; dn[c][4] = e.w;
            dn[c][5] = f.x; dn[c][6] = f.y; dn[c][7] = f.z; dn[c][8] = f.w;
            dn[c][0] = s_ch[c][o + WW - 1];               // validity enforced at use
            dn[c][9] = s_ch[c][o + WW + 8];               // validity enforced at use
        }
    }

    float acc = 0.0f;

    #pragma unroll
    for (int i = 0; i < 8; ++i) {
        const int w = w0 + i;
        const float a0 = cur[0][i], a1 = cur[1][i], a2 = cur[2][i], yv = cur[3][i];

        // shift (0,+1): neighbor cur[*][i+1]
        if (w + 1 < WW) {
            float d0 = a0 - cur[0][i + 1], d1 = a1 - cur[1][i + 1], d2 = a2 - cur[2][i + 1];
            float dy = yv - cur[3][i + 1];
            float m  = fast_exp2((d0 * d0 + d1 * d1 + d2 * d2) * kExp2Scale) * kW10 - 0.01f;
            acc += m * (dy * dy);
        }
        if (below) {
            // shift (+1,0): dn[*][i+1]
            {
                float d0 = a0 - dn[0][i + 1], d1 = a1 - dn[1][i + 1], d2 = a2 - dn[2][i + 1];
                float dy = yv - dn[3][i + 1];
                float m  = fast_exp2((d0 * d0 + d1 * d1 + d2 * d2) * kExp2Scale) * kW10 - 0.01f;
                acc += m * (dy * dy);
            }
            // shift (+1,+1): dn[*][i+2]
            if (w + 1 < WW) {
                float d0 = a0 - dn[0][i + 2], d1 = a1 - dn[1][i + 2], d2 = a2 - dn[2][i + 2];
                float dy = yv - dn[3][i + 2];
                float m  = fast_exp2((d0 * d0 + d1 * d1 + d2 * d2) * kExp2Scale) * kW11 - 0.01f;
                acc += m * (dy * dy);
            }
            // shift (+1,-1): dn[*][i]
            if (w >= 1) {
                float d0 = a0 - dn[0][i], d1 = a1 - dn[1][i], d2 = a2 - dn[2][i];
                float dy = yv - dn[3][i];
                float m  = fast_exp2((d0 * d0 + d1 * d1 + d2 * d2) * kExp2Scale) * kW11 - 0.01f;
                acc += m * (dy * dy);
            }
        }
    }

    // Fold WEIGHT * (1/(H*W)) * (1/B) * (1/len(SHIFTS)) = 2^-24.
    acc *= 5.9604644775390625e-08f;

    // wave32 shuffle reduction -> 8 partials -> final shuffle -> one atomic.
    #pragma unroll
    for (int off = 16; off > 0; off >>= 1)
        acc += __shfl_down(acc, off, 32);
    const int wave = tid >> 5, lane = tid & 31;
    if (lane == 0) s_red[wave] = acc;
    __syncthreads();
    if (tid < 8) {
        float t = s_red[tid];
        #pragma unroll
        for (int off = 4; off > 0; off >>= 1)
            t += __shfl_down(t, off, 32);   // lanes >=4 carry junk, never consumed
        if (tid == 0) unsafeAtomicAdd(out, t);
    }
}

extern "C" void kernel_launch(void* const* d_in, const int* in_sizes, int n_in,
                              void* d_out, int out_size, void* d_ws, size_t ws_size,
                              hipStream_t stream) {
    const float* y_pr        = (const float*)d_in[0];
    // d_in[1] = y_gt (int32) — unused by the reference loss.
    const float* image       = (const float*)d_in[2];
    const int*   image_class = (const int*)d_in[3];
    float*       out         = (float*)d_out;

    crf_zero_kernel<<<1, 1, 0, stream>>>(out);

    const int grid = BB * (HH / ROWS);   // 16 * 128 = 2048 blocks
    crf_loss_kernel<<<grid, THREADS, 0, stream>>>(y_pr, image, image_class, out);
}